// SELayer_40699110096955
// MI455X (gfx1250) — compile-verified
//
#include <hip/hip_runtime.h>
#include <hip/hip_bf16.h>
#include <math.h>

typedef __attribute__((ext_vector_type(2))) float v2f;
typedef __attribute__((ext_vector_type(8))) float v8f;

#define NB 8     // batches
#define C  256   // channels
#define CR 64    // channels / reduction

// ---------------------------------------------------------------------------
// Kernel 1: deterministic per-block partial segment sums.
// Each block owns a contiguous row chunk; thread c owns channel c exclusively
// in the 8x256 LDS accumulator, so no atomics and a fixed summation order.
// ---------------------------------------------------------------------------
__global__ __launch_bounds__(256) void se_partial_kernel(
    const float* __restrict__ x, const int* __restrict__ bids,
    int N, int rows_per_blk,
    float* __restrict__ partial,   // [P][8][256]
    float* __restrict__ pcnt)      // [P][8]
{
    __shared__ float acc[NB * C];
    __shared__ float cnt[NB];
    const int c = threadIdx.x;
#pragma unroll
    for (int b = 0; b < NB; ++b) acc[b * C + c] = 0.0f;
    if (c < NB) cnt[c] = 0.0f;
    __syncthreads();

    int r0 = blockIdx.x * rows_per_blk;
    int r1 = r0 + rows_per_blk;
    if (r1 > N) r1 = N;
    for (int r = r0; r < r1; ++r) {
        const int b = bids[r];                      // uniform across block
        acc[b * C + c] += x[(size_t)r * C + c];     // coalesced 1KB/row
        if (c == 0) cnt[b] += 1.0f;
    }
    __syncthreads();

    float* pp = partial + (size_t)blockIdx.x * (NB * C);
#pragma unroll
    for (int b = 0; b < NB; ++b) pp[b * C + c] = acc[b * C + c];
    if (c < NB) pcnt[blockIdx.x * NB + c] = cnt[c];
}

// ---------------------------------------------------------------------------
// Kernel 2: reduce partials in fixed order, divide by counts, pad to 16 rows.
// ---------------------------------------------------------------------------
__global__ __launch_bounds__(256) void se_reduce_kernel(
    const float* __restrict__ partial, const float* __restrict__ pcnt,
    int P, float* __restrict__ pooled /* [16][256], rows 8..15 zeroed */)
{
    const int t = threadIdx.x;
    __shared__ float scnt[NB];
    if (t < NB) {
        float s = 0.0f;
        for (int p = 0; p < P; ++p) s += pcnt[p * NB + t];
        scnt[t] = s > 1.0f ? s : 1.0f;
    }
    __syncthreads();
    for (int idx = t; idx < NB * C; idx += 256) {
        const int b = idx >> 8;
        float s = 0.0f;
        for (int p = 0; p < P; ++p) s += partial[(size_t)p * (NB * C) + idx];
        pooled[idx] = s / scnt[b];
    }
    for (int idx = NB * C + t; idx < 16 * C; idx += 256) pooled[idx] = 0.0f;
}

// ---------------------------------------------------------------------------
// Kernel 3: squeeze-excite MLP on one wave32 with V_WMMA_F32_16X16X4_F32.
// A(16x4 f32): lane l<16 holds {K=k0, K=k0+1} of row M=l, lanes 16..31 hold
// {K=k0+2, K=k0+3}. B(4x16) mirrors (K striped VGPR-first, lane-half second).
// D VGPR r -> row M = r + 8*(lane>>4), col N = lane&15.
// ---------------------------------------------------------------------------
__global__ __launch_bounds__(32) void se_mlp_kernel(
    const float* __restrict__ pooled,   // [16][256]
    const float* __restrict__ W1,       // [256][64]
    const float* __restrict__ b1,       // [64]
    const float* __restrict__ W2,       // [64][256]
    const float* __restrict__ b2,       // [256]
    float* __restrict__ y)              // [16][256]
{
    __shared__ float Hs[16 * CR];
    const int lane = threadIdx.x;
    const int half = lane >> 4;
    const int ln   = lane & 15;

    // GEMM1: H = relu(pooled @ W1 + b1)  -> [16,64]
    for (int n0 = 0; n0 < CR; n0 += 16) {
        v8f acc = {};
        for (int k0 = 0; k0 < C; k0 += 4) {
            const int ka = k0 + 2 * half;
            v2f a, b;
            a[0] = pooled[ln * C + ka];
            a[1] = pooled[ln * C + ka + 1];
            b[0] = W1[ka * CR + n0 + ln];
            b[1] = W1[(ka + 1) * CR + n0 + ln];
            acc = __builtin_amdgcn_wmma_f32_16x16x4_f32(
                false, a, false, b, (short)0, acc, false, false);
        }
        const float bias = b1[n0 + ln];
#pragma unroll
        for (int r = 0; r < 8; ++r) {
            const int m = r + 8 * half;
            float h = acc[r] + bias;
            Hs[m * CR + n0 + ln] = h > 0.0f ? h : 0.0f;
        }
    }
    __syncthreads();

    // GEMM2: Y = sigmoid(H @ W2 + b2)  -> [16,256]
    for (int n0 = 0; n0 < C; n0 += 16) {
        v8f acc = {};
        for (int k0 = 0; k0 < CR; k0 += 4) {
            const int ka = k0 + 2 * half;
            v2f a, b;
            a[0] = Hs[ln * CR + ka];
            a[1] = Hs[ln * CR + ka + 1];
            b[0] = W2[ka * C + n0 + ln];
            b[1] = W2[(ka + 1) * C + n0 + ln];
            acc = __builtin_amdgcn_wmma_f32_16x16x4_f32(
                false, a, false, b, (short)0, acc, false, false);
        }
        const float bias = b2[n0 + ln];
#pragma unroll
        for (int r = 0; r < 8; ++r) {
            const int m = r + 8 * half;
            const float v = acc[r] + bias;
            y[m * C + n0 + ln] = 1.0f / (1.0f + expf(-v));
        }
    }
}

// ---------------------------------------------------------------------------
// Kernel 4: out[n,c] = y[bids[n], c] * x[n, c], float4-vectorized.
// 512 MB of streaming traffic -> the bandwidth-critical kernel.
// ---------------------------------------------------------------------------
__global__ __launch_bounds__(256) void se_scale_kernel(
    const float4* __restrict__ x4, const int* __restrict__ bids,
    const float* __restrict__ y, float4* __restrict__ out4, long long total4)
{
    const long long stride = (long long)gridDim.x * blockDim.x;
    const float4* y4 = (const float4*)y;
    for (long long i = (long long)blockIdx.x * blockDim.x + threadIdx.x;
         i < total4; i += stride) {
        const int row = (int)(i >> 6);     // C/4 = 64 float4 per row
        const int q   = (int)(i & 63);
        const int b   = bids[row];
        const float4 xv = x4[i];
        const float4 yv = y4[b * 64 + q];  // 8KB table, L2-resident
        float4 o;
        o.x = xv.x * yv.x; o.y = xv.y * yv.y;
        o.z = xv.z * yv.z; o.w = xv.w * yv.w;
        out4[i] = o;
    }
}

// ---------------------------------------------------------------------------
extern "C" void kernel_launch(void* const* d_in, const int* in_sizes, int n_in,
                              void* d_out, int out_size, void* d_ws, size_t ws_size,
                              hipStream_t stream) {
    const float* x    = (const float*)d_in[0];
    const int*   bids = (const int*)d_in[1];
    const float* W1   = (const float*)d_in[2];
    const float* b1   = (const float*)d_in[3];
    const float* W2   = (const float*)d_in[4];
    const float* b2   = (const float*)d_in[5];
    float*       out  = (float*)d_out;
    const int N = in_sizes[0] / C;

    // Workspace layout: pooled[16*256] | y[16*256] | pcnt[P*8] | partial[P*8*256]
    float* pooled = (float*)d_ws;
    float* yb     = pooled + 16 * C;
    float* pcnt   = yb + 16 * C;
    const size_t fixed_bytes = (size_t)(2 * 16 * C) * sizeof(float);
    const size_t per_block   = (size_t)(NB * C + NB) * sizeof(float);
    int P = 512;
    if (ws_size > fixed_bytes) {
        const size_t avail = (ws_size - fixed_bytes) / per_block;
        if ((size_t)P > avail) P = (int)avail;
    } else {
        P = 1;
    }
    if (P < 1) P = 1;
    float* partial = pcnt + (size_t)P * NB;
    const int rows_per_blk = (N + P - 1) / P;

    se_partial_kernel<<<P, 256, 0, stream>>>(x, bids, N, rows_per_blk, partial, pcnt);
    se_reduce_kernel<<<1, 256, 0, stream>>>(partial, pcnt, P, pooled);
    se_mlp_kernel<<<1, 32, 0, stream>>>(pooled, W1, b1, W2, b2, yb);

    const long long total4 = (long long)N * (C / 4);
    long long need = (total4 + 255) / 256;
    int blocks4 = need < 8192 ? (int)need : 8192;
    if (blocks4 < 1) blocks4 = 1;
    se_scale_kernel<<<blocks4, 256, 0, stream>>>(
        (const float4*)x, bids, yb, (float4*)out, total4);
}